// TranscriptEmission_16664473108524
// MI455X (gfx1250) — compile-verified
//
#include <hip/hip_runtime.h>

// TranscriptEmission: out[s, i] = LUT[s][obs[i]]  (pure lookup)
//   row 0: obs==0 ? 0 : -100000
//   rows 1..24: sum_{k<x} log(phi+k) - log(x!) + phi*log(phi/(phi+mu)) + x*log(mu/(phi+mu))
// Store-bandwidth bound: 400 MB NT stores @ 23.3 TB/s ~= 18 us floor.

#define N_STATES   25
#define LUT_STRIDE 64     // padded row stride (floats); distinct x -> distinct LDS bank
#define LOGP_PREC  -100000.0f

typedef __attribute__((ext_vector_type(4))) float v4f;
typedef __attribute__((ext_vector_type(4))) int   v4i;

__global__ __launch_bounds__(256)
void emission_lut_kernel(const float* __restrict__ means,
                         const float* __restrict__ phis,
                         const int*   __restrict__ obs,
                         float*       __restrict__ out,
                         int n_spots)
{
    __shared__ float lut[N_STATES * LUT_STRIDE];

    const int tid = threadIdx.x;

    // ---- Build the 25x64 LUT once per block (one lane per state) ----
    if (tid < N_STATES) {
        float* row = &lut[tid * LUT_STRIDE];
        if (tid == 0) {
            row[0] = 0.0f;
            #pragma unroll 1
            for (int x = 1; x < LUT_STRIDE; ++x) row[x] = LOGP_PREC;
        } else {
            const float mu  = means[tid];
            const float phi = phis[tid];
            const float inv = 1.0f / (phi + mu);
            const float c1  = phi * logf(phi * inv);   // phi*log(phi/(phi+mu))
            const float c2  = logf(mu * inv);          // log(mu/(phi+mu))
            float acc = 0.0f;   // sum_{k<x} log(phi+k) == gammaln(x+phi)-gammaln(phi)
            float lf  = 0.0f;   // log(x!)              == gammaln(x+1)
            #pragma unroll 1
            for (int x = 0; x < LUT_STRIDE; ++x) {
                row[x] = acc - lf + c1 + (float)x * c2;
                acc += logf(phi + (float)x);
                lf  += logf((float)(x + 1));
            }
        }
    }
    __syncthreads();

    const long long N  = n_spots;
    const int nthreads = (int)(gridDim.x * blockDim.x);
    const int gtid     = (int)(blockIdx.x * blockDim.x) + tid;

    // ---- Vector main loop: 4 spots/thread/iter, 25 NT b128 stores ----
    const int nvec = ((n_spots & 3) == 0) ? (n_spots >> 2) : 0;
    const v4i* __restrict__ obs4 = (const v4i*)obs;

    for (int i = gtid; i < nvec; i += nthreads) {
        // prefetch next grid-stride iteration's obs vector (speculative; OOB dropped)
        __builtin_prefetch((const void*)(obs4 + i + nthreads), 0, 1);

        const v4i o = __builtin_nontemporal_load(obs4 + i);
        const int x0 = o.x & (LUT_STRIDE - 1);
        const int x1 = o.y & (LUT_STRIDE - 1);
        const int x2 = o.z & (LUT_STRIDE - 1);
        const int x3 = o.w & (LUT_STRIDE - 1);

        float* base = out + (((long long)i) << 2);   // column base; row step = N
        #pragma unroll
        for (int s = 0; s < N_STATES; ++s) {
            const float* __restrict__ row = &lut[s * LUT_STRIDE];
            v4f v = { row[x0], row[x1], row[x2], row[x3] };
            __builtin_nontemporal_store(v, (v4f*)(base + (long long)s * N));
        }
    }

    // ---- Scalar tail (covers everything if n_spots % 4 != 0) ----
    for (int i = (nvec << 2) + gtid; i < n_spots; i += nthreads) {
        const int x = obs[i] & (LUT_STRIDE - 1);
        #pragma unroll
        for (int s = 0; s < N_STATES; ++s) {
            __builtin_nontemporal_store(lut[s * LUT_STRIDE + x],
                                        out + (long long)s * N + i);
        }
    }
}

extern "C" void kernel_launch(void* const* d_in, const int* in_sizes, int n_in,
                              void* d_out, int out_size, void* d_ws, size_t ws_size,
                              hipStream_t stream) {
    const float* means = (const float*)d_in[0];  // state_means[25]
    const float* phis  = (const float*)d_in[1];  // state_phis[25]
    const int*   obs   = (const int*)  d_in[2];  // obs[n]
    float*       out   = (float*)d_out;          // [25, n] row-major
    const int n = in_sizes[2];

    const int work = ((n & 3) == 0) ? (n >> 2) : n;  // vec-groups or scalar count
    int blocks = (work + 255) / 256;
    // Cap low enough that each thread grid-strides ~4 iterations: amortizes the
    // per-block LDS LUT build 4x while 1024 blocks x 8 wave32 = 8192 waves still
    // vastly oversubscribes the WGPs needed to saturate 23.3 TB/s of NT stores.
    if (blocks > 1024) blocks = 1024;
    if (blocks < 1)    blocks = 1;

    emission_lut_kernel<<<blocks, 256, 0, stream>>>(means, phis, obs, out, n);
}